// MNN_42116449304834
// MI455X (gfx1250) — compile-verified
//
#include <hip/hip_runtime.h>
#include <math.h>

// ---------------------------------------------------------------------------
// Problem constants
// ---------------------------------------------------------------------------
#define N_MAN 12        // manifold dim
#define V_DIM 78        // N*(N+1)/2
#define V_PAD 80        // padded to 5x16 row tiles
#define H_DIM 128
#define B_TOT 2048
#define BT    16        // batch tile per workgroup in MLP kernel

typedef float v2f __attribute__((ext_vector_type(2)));
typedef float v8f __attribute__((ext_vector_type(8)));

#if defined(__has_builtin)
#if __has_builtin(__builtin_amdgcn_global_load_async_to_lds_b32) && \
    __has_builtin(__builtin_amdgcn_s_wait_asynccnt)
#define USE_ASYNC_LDS 1
#endif
#endif

// workspace layout (in floats); weights stored pre-packed in WMMA A-fragment
// order: Wp[(r0*Ksteps + ks)*64 + lane*2 + j]
#define W1_STEPS 4
#define W2_STEPS 32
#define W3_STEPS 32
#define OFF_W1P 0                                   // 8*4*64   = 2048
#define OFF_W2P (OFF_W1P + 8 * W1_STEPS * 64)       // 8*32*64  = 16384
#define OFF_W3P (OFF_W2P + 8 * W2_STEPS * 64)       // 5*32*64  = 10240
#define OFF_UV  (OFF_W3P + 5 * W3_STEPS * 64)       // [2048][80]
#define OFF_DUV (OFF_UV + B_TOT * V_PAD)            // [2048][80][12]

// ---------------------------------------------------------------------------
// CDNA5 f32 WMMA 16x16x4 : D = A(16x4) * B(4x16) + C
// ---------------------------------------------------------------------------
__device__ __forceinline__ v8f wmma4(v2f a, v2f b, v8f c) {
  return __builtin_amdgcn_wmma_f32_16x16x4_f32(
      /*neg_a=*/false, a, /*neg_b=*/false, b,
      /*c_mod=*/(short)0, c, /*reuse_a=*/false, /*reuse_b=*/false);
}

__device__ __forceinline__ float softplus_f(float x) {
  return fmaxf(x, 0.f) + log1pf(expf(-fabsf(x)));   // branchless, stable
}
__device__ __forceinline__ float sigmoid_f(float x) {
  return 1.f / (1.f + expf(-x));
}

// One 16x16 output tile of  Y = W * Z  with pre-packed A fragments.
// Wp: per-lane fragment order (one b64 load per WMMA). Z in LDS: Z[k*16 + n].
__device__ __forceinline__ v8f gemm_tile_p(const float* __restrict__ Wp,
                                           const float* __restrict__ Z,
                                           int r0, int ksteps, int lane) {
  const int nn = lane & 15;
  const int ko = (lane >> 4) << 1;
  const float* wp = Wp + (r0 * ksteps) * 64 + lane * 2;
  v8f acc = {0.f, 0.f, 0.f, 0.f, 0.f, 0.f, 0.f, 0.f};
#pragma unroll 4
  for (int ks = 0; ks < ksteps; ++ks) {
    v2f a = *(const v2f*)(wp + ks * 64);
    const int kA = ks * 4 + ko;
    v2f b;
    b.x = Z[(kA + 0) * 16 + nn];
    b.y = Z[(kA + 1) * 16 + nn];
    acc = wmma4(a, b, acc);
  }
  return acc;
}

// ---------------------------------------------------------------------------
// Kernel A: pack weights into per-lane WMMA A-fragment order (zero padded).
// ---------------------------------------------------------------------------
__device__ __forceinline__ void pack_frag(const float* __restrict__ W,
                                          float* __restrict__ Wp,
                                          int Mtiles, int Ksteps, int Kdim,
                                          int Mdim, int ldW, int tid, int nth) {
  const int total = Mtiles * Ksteps * 64;
  for (int i = tid; i < total; i += nth) {
    int j    = i & 1;
    int lane = (i >> 1) & 31;
    int ks   = (i >> 6) % Ksteps;
    int r0   = (i >> 6) / Ksteps;
    int m = r0 * 16 + (lane & 15);
    int k = ks * 4 + ((lane >> 4) << 1) + j;
    Wp[i] = (k < Kdim && m < Mdim) ? W[m * ldW + k] : 0.f;
  }
}

__global__ void mnn_prep_kernel(const float* __restrict__ W1,
                                const float* __restrict__ W2,
                                const float* __restrict__ W3,
                                float* __restrict__ ws) {
  const int tid = blockIdx.x * blockDim.x + threadIdx.x;
  const int nth = gridDim.x * blockDim.x;
  pack_frag(W1, ws + OFF_W1P, 8, W1_STEPS, N_MAN, H_DIM, N_MAN, tid, nth);
  pack_frag(W2, ws + OFF_W2P, 8, W2_STEPS, H_DIM, H_DIM, H_DIM, tid, nth);
  pack_frag(W3, ws + OFF_W3P, 5, W3_STEPS, H_DIM, V_DIM, H_DIM, tid, nth);
}

// ---------------------------------------------------------------------------
// Kernel B: differentiated MLP via f32 WMMA.
// One WG = 128 threads (4 waves) handles 16 batch elements.
// ---------------------------------------------------------------------------
__global__ __launch_bounds__(128) void mnn_mlp_kernel(
    const float* __restrict__ q, const float* __restrict__ dqdq,
    const float* __restrict__ b1, const float* __restrict__ b2,
    const float* __restrict__ b3, float* __restrict__ ws) {
  __shared__ float sAct [H_DIM * BT];   // current value activations
  __shared__ float sSig1[H_DIM * BT];   // sigmoid(pre1)
  __shared__ float sSig2[H_DIM * BT];   // sigmoid(pre2)
  __shared__ float sD0  [H_DIM * BT];   // ping
  __shared__ float sD1  [H_DIM * BT];   // pong

  const float* W1p = ws + OFF_W1P;
  const float* W2p = ws + OFF_W2P;
  const float* W3p = ws + OFF_W3P;
  float* Uv  = ws + OFF_UV;
  float* dUv = ws + OFF_DUV;

  const int tid  = threadIdx.x;
  const int lane = tid & 31;
  const int wave = tid >> 5;
  const int b0   = blockIdx.x * BT;

  const int n  = lane & 15;          // tile column = batch within tile
  const int mo = (lane >> 4) << 3;   // +8 row offset for upper half-wave

  // Pull packed W2 toward this WGP early (global_prefetch_b8).
  __builtin_prefetch(W2p + tid * H_DIM, 0, 0);

  // ---- stage q tile into sD0 as K=16 x 16 (rows 12..15 zero) ----
#ifdef USE_ASYNC_LDS
  if (tid < 4 * BT) sD0[12 * BT + tid] = 0.f;     // zero pad rows 12..15
  for (int e = tid; e < 12 * BT; e += 128) {
    int k = e >> 4, c = e & 15;
    const float* g = q + (b0 + c) * N_MAN + k;
    __builtin_amdgcn_global_load_async_to_lds_b32((int*)g, (int*)&sD0[e], 0, 0);
  }
  __builtin_amdgcn_s_wait_asynccnt(0);
#else
  for (int e = tid; e < 16 * BT; e += 128) {
    int k = e >> 4, c = e & 15;
    sD0[e] = (k < N_MAN) ? q[(b0 + c) * N_MAN + k] : 0.f;
  }
#endif
  __syncthreads();

  // ---- Layer 1 values ----
  {
    v8f a0 = gemm_tile_p(W1p, sD0, wave,     W1_STEPS, lane);
    v8f a1 = gemm_tile_p(W1p, sD0, wave + 4, W1_STEPS, lane);
#pragma unroll
    for (int v = 0; v < 8; ++v) {
      int m0 = wave * 16 + mo + v, m1 = (wave + 4) * 16 + mo + v;
      float x0 = a0[v] + b1[m0], x1 = a1[v] + b1[m1];
      sSig1[m0 * 16 + n] = sigmoid_f(x0);
      sAct [m0 * 16 + n] = softplus_f(x0);
      sSig1[m1 * 16 + n] = sigmoid_f(x1);
      sAct [m1 * 16 + n] = softplus_f(x1);
    }
  }
  __syncthreads();

  // ---- Layer 2 values ----
  {
    v8f a0 = gemm_tile_p(W2p, sAct, wave,     W2_STEPS, lane);
    v8f a1 = gemm_tile_p(W2p, sAct, wave + 4, W2_STEPS, lane);
    __syncthreads();   // all reads of sAct complete before overwrite
#pragma unroll
    for (int v = 0; v < 8; ++v) {
      int m0 = wave * 16 + mo + v, m1 = (wave + 4) * 16 + mo + v;
      float x0 = a0[v] + b2[m0], x1 = a1[v] + b2[m1];
      sSig2[m0 * 16 + n] = sigmoid_f(x0);
      sAct [m0 * 16 + n] = softplus_f(x0);
      sSig2[m1 * 16 + n] = sigmoid_f(x1);
      sAct [m1 * 16 + n] = softplus_f(x1);
    }
  }
  __syncthreads();

  // ---- Layer 3 values -> Uvect (workspace) ----
  {
    v8f a0 = gemm_tile_p(W3p, sAct, wave, W3_STEPS, lane);
#pragma unroll
    for (int v = 0; v < 8; ++v) {
      int m = wave * 16 + mo + v;
      if (m < V_DIM) Uv[(b0 + n) * V_PAD + m] = a0[v] + b3[m];
    }
    if (wave == 0) {   // wave-uniform branch: EXEC stays all-ones in-wave
      v8f a1 = gemm_tile_p(W3p, sAct, 4, W3_STEPS, lane);
#pragma unroll
      for (int v = 0; v < 8; ++v) {
        int m = 64 + mo + v;
        if (m < V_DIM) Uv[(b0 + n) * V_PAD + m] = a1[v] + b3[m];
      }
    }
  }

  // ---- Tangent passes: one per manifold direction l ----
  for (int l = 0; l < N_MAN; ++l) {
    __syncthreads();   // previous D3 reads of sD0 complete
#ifdef USE_ASYNC_LDS
    if (tid < 4 * BT) sD0[12 * BT + tid] = 0.f;
    for (int e = tid; e < 12 * BT; e += 128) {
      int k = e >> 4, c = e & 15;
      const float* g = dqdq + ((b0 + c) * N_MAN + k) * N_MAN + l;
      __builtin_amdgcn_global_load_async_to_lds_b32((int*)g, (int*)&sD0[e], 0, 0);
    }
    __builtin_amdgcn_s_wait_asynccnt(0);
#else
    for (int e = tid; e < 16 * BT; e += 128) {
      int k = e >> 4, c = e & 15;
      sD0[e] = (k < N_MAN) ? dqdq[((b0 + c) * N_MAN + k) * N_MAN + l] : 0.f;
    }
#endif
    __syncthreads();

    // D1 = sig1 .* (W1 * S)
    {
      v8f a0 = gemm_tile_p(W1p, sD0, wave,     W1_STEPS, lane);
      v8f a1 = gemm_tile_p(W1p, sD0, wave + 4, W1_STEPS, lane);
#pragma unroll
      for (int v = 0; v < 8; ++v) {
        int m0 = wave * 16 + mo + v, m1 = (wave + 4) * 16 + mo + v;
        sD1[m0 * 16 + n] = a0[v] * sSig1[m0 * 16 + n];
        sD1[m1 * 16 + n] = a1[v] * sSig1[m1 * 16 + n];
      }
    }
    __syncthreads();

    // D2 = sig2 .* (W2 * D1)   (into sD0)
    {
      v8f a0 = gemm_tile_p(W2p, sD1, wave,     W2_STEPS, lane);
      v8f a1 = gemm_tile_p(W2p, sD1, wave + 4, W2_STEPS, lane);
#pragma unroll
      for (int v = 0; v < 8; ++v) {
        int m0 = wave * 16 + mo + v, m1 = (wave + 4) * 16 + mo + v;
        sD0[m0 * 16 + n] = a0[v] * sSig2[m0 * 16 + n];
        sD0[m1 * 16 + n] = a1[v] * sSig2[m1 * 16 + n];
      }
    }
    __syncthreads();

    // D3 = W3 * D2 -> dUvect (workspace)
    {
      v8f a0 = gemm_tile_p(W3p, sD0, wave, W3_STEPS, lane);
#pragma unroll
      for (int v = 0; v < 8; ++v) {
        int m = wave * 16 + mo + v;
        if (m < V_DIM) dUv[((b0 + n) * V_PAD + m) * N_MAN + l] = a0[v];
      }
      if (wave == 0) {
        v8f a1 = gemm_tile_p(W3p, sD0, 4, W3_STEPS, lane);
#pragma unroll
        for (int v = 0; v < 8; ++v) {
          int m = 64 + mo + v;
          if (m < V_DIM) dUv[((b0 + n) * V_PAD + m) * N_MAN + l] = a1[v];
        }
      }
    }
  }
}

// ---------------------------------------------------------------------------
// Kernel C: per-batch 12x12 symmetric eigendecomposition (parallel Jacobi),
// M = Q e^L Q^T, and dM/dq via Daleckii-Krein:  dM_l = Q (F .* (Q^T S_l Q)) Q^T
// One wave per matrix, 4 matrices per workgroup.
// ---------------------------------------------------------------------------
#define NMAT 4

__device__ __forceinline__ void tril_rc(int k, int& r, int& c) {
  r = (int)((sqrtf(8.f * (float)k + 1.f) - 1.f) * 0.5f);
  while ((r + 1) * (r + 2) / 2 <= k) ++r;
  while (r * (r + 1) / 2 > k) --r;
  c = k - r * (r + 1) / 2;
}

__global__ __launch_bounds__(128) void mnn_eig_kernel(const float* __restrict__ ws,
                                                      float* __restrict__ out) {
  __shared__ float sA[NMAT][144];
  __shared__ float sQ[NMAT][144];
  __shared__ float sT[NMAT][144];
  __shared__ float sS[NMAT][144];
  __shared__ float sF[NMAT][144];
  __shared__ float sw[NMAT][12];
  __shared__ float sew[NMAT][12];
  __shared__ float scc[NMAT][12];
  __shared__ float sss[NMAT][12];
  __shared__ int   sprt[NMAT][12];
  __shared__ int   sisp[NMAT][12];

  const int lane = threadIdx.x & 31;
  const int mat  = threadIdx.x >> 5;
  const int b    = blockIdx.x * NMAT + mat;

  float* A  = sA[mat];  float* Q  = sQ[mat];  float* T = sT[mat];
  float* S  = sS[mat];  float* F  = sF[mat];
  float* w  = sw[mat];  float* ew = sew[mat];
  float* cc = scc[mat]; float* ssn = sss[mat];
  int*   prt = sprt[mat]; int* isp = sisp[mat];

  const float* Uv  = ws + OFF_UV  + (size_t)b * V_PAD;
  const float* dUv = ws + OFF_DUV + (size_t)b * V_PAD * N_MAN;

  // init Q = I, load symmetric A from Uvect (tril packing)
  for (int e = lane; e < 144; e += 32) Q[e] = ((e / 12) == (e % 12)) ? 1.f : 0.f;
  for (int k = lane; k < V_DIM; k += 32) {
    int r, c; tril_rc(k, r, c);
    float v = Uv[k];
    A[r * 12 + c] = v;
    A[c * 12 + r] = v;
  }
  __syncthreads();

  // parallel cyclic Jacobi: 10 sweeps x 11 rounds, 6 disjoint rotations/round.
  // Eigenvectors ping-pong between Q and S (110 rounds: even -> ends in Q).
  float* Qa = Q;
  float* Qb = S;
  for (int sweep = 0; sweep < 10; ++sweep) {
    for (int rr = 0; rr < 11; ++rr) {
      if (lane < 6) {
        int p, qq;
        if (lane == 0) { p = 11; qq = rr % 11; }
        else { p = (rr + lane) % 11; qq = (rr + 11 - lane) % 11; }
        float app = A[p * 12 + p], aqq = A[qq * 12 + qq], apq = A[p * 12 + qq];
        float c, s;
        if (fabsf(apq) < 1e-30f) { c = 1.f; s = 0.f; }
        else {
          float tau = (aqq - app) / (2.f * apq);
          float t = ((tau >= 0.f) ? 1.f : -1.f) / (fabsf(tau) + sqrtf(1.f + tau * tau));
          c = 1.f / sqrtf(1.f + t * t);
          s = t * c;
        }
        prt[p] = qq; prt[qq] = p;
        cc[p] = c;  cc[qq] = c;
        ssn[p] = s; ssn[qq] = s;
        isp[p] = 1; isp[qq] = 0;
      }
      __syncthreads();
      // T = J^T A   (row mix)
      for (int e = lane; e < 144; e += 32) {
        int i = e / 12, j = e - i * 12;
        int pi = prt[i]; float ci = cc[i], si = ssn[i];
        T[e] = isp[i] ? (ci * A[i * 12 + j] - si * A[pi * 12 + j])
                      : (si * A[pi * 12 + j] + ci * A[i * 12 + j]);
      }
      __syncthreads();
      // A = T J,  Qb = Qa J   (column mix)
      for (int e = lane; e < 144; e += 32) {
        int i = e / 12, j = e - i * 12;
        int pj = prt[j]; float cj = cc[j], sj = ssn[j];
        A[e]  = isp[j] ? (cj * T[i * 12 + j] - sj * T[i * 12 + pj])
                       : (sj * T[i * 12 + pj] + cj * T[i * 12 + j]);
        Qb[e] = isp[j] ? (cj * Qa[i * 12 + j] - sj * Qa[i * 12 + pj])
                       : (sj * Qa[i * 12 + pj] + cj * Qa[i * 12 + j]);
      }
      __syncthreads();
      float* tmp = Qa; Qa = Qb; Qb = tmp;
    }
  }
  // 110 rounds (even): latest eigenvectors are back in Q; S is free.

  if (lane < 12) {
    float wi = A[lane * 12 + lane];
    w[lane]  = wi;
    ew[lane] = expf(wi);
  }
  __syncthreads();

  // Loewner matrix F and output M = Q diag(ew) Q^T
  for (int e = lane; e < 144; e += 32) {
    int i = e / 12, j = e - i * 12;
    float dw = w[i] - w[j];
    F[e] = (fabsf(dw) > 1e-5f) ? (ew[i] - ew[j]) / dw
                               : expf(0.5f * (w[i] + w[j]));
  }
  for (int e = lane; e < 144; e += 32) {
    int i = e / 12, j = e - i * 12;
    float acc = 0.f;
    for (int k = 0; k < 12; ++k) acc += Q[i * 12 + k] * ew[k] * Q[j * 12 + k];
    out[(size_t)b * 144 + e] = acc;
  }
  __syncthreads();

  float* dM = out + (size_t)B_TOT * 144;
  for (int l = 0; l < N_MAN; ++l) {
    // S_l from dUvect[:, l] (symmetric scatter)
    for (int k = lane; k < V_DIM; k += 32) {
      int r, c; tril_rc(k, r, c);
      float v = dUv[k * N_MAN + l];
      S[r * 12 + c] = v;
      S[c * 12 + r] = v;
    }
    __syncthreads();
    // T = Q^T S
    for (int e = lane; e < 144; e += 32) {
      int i = e / 12, j = e - i * 12;
      float acc = 0.f;
      for (int k = 0; k < 12; ++k) acc += Q[k * 12 + i] * S[k * 12 + j];
      T[e] = acc;
    }
    __syncthreads();
    // S = F .* (T Q)
    for (int e = lane; e < 144; e += 32) {
      int i = e / 12, j = e - i * 12;
      float acc = 0.f;
      for (int k = 0; k < 12; ++k) acc += T[i * 12 + k] * Q[k * 12 + j];
      S[e] = acc * F[e];
    }
    __syncthreads();
    // T = Q S
    for (int e = lane; e < 144; e += 32) {
      int i = e / 12, j = e - i * 12;
      float acc = 0.f;
      for (int k = 0; k < 12; ++k) acc += Q[i * 12 + k] * S[k * 12 + j];
      T[e] = acc;
    }
    __syncthreads();
    // dM[:, :, l] = T Q^T
    for (int e = lane; e < 144; e += 32) {
      int i = e / 12, j = e - i * 12;
      float acc = 0.f;
      for (int k = 0; k < 12; ++k) acc += T[i * 12 + k] * Q[j * 12 + k];
      dM[((size_t)b * 144 + e) * N_MAN + l] = acc;
    }
    __syncthreads();
  }
}

// ---------------------------------------------------------------------------
// Launch
// ---------------------------------------------------------------------------
extern "C" void kernel_launch(void* const* d_in, const int* in_sizes, int n_in,
                              void* d_out, int out_size, void* d_ws, size_t ws_size,
                              hipStream_t stream) {
  (void)in_sizes; (void)n_in; (void)out_size; (void)ws_size;
  const float* q    = (const float*)d_in[0];
  const float* dqdq = (const float*)d_in[1];
  const float* W1   = (const float*)d_in[2];
  const float* b1   = (const float*)d_in[3];
  const float* W2   = (const float*)d_in[4];
  const float* b2   = (const float*)d_in[5];
  const float* W3   = (const float*)d_in[6];
  const float* b3   = (const float*)d_in[7];
  float* ws  = (float*)d_ws;
  float* out = (float*)d_out;

  mnn_prep_kernel<<<32, 256, 0, stream>>>(W1, W2, W3, ws);
  mnn_mlp_kernel<<<B_TOT / BT, 128, 0, stream>>>(q, dqdq, b1, b2, b3, ws);
  mnn_eig_kernel<<<B_TOT / NMAT, 128, 0, stream>>>(ws, out);
}